// ChamferMse_55886114456074
// MI455X (gfx1250) — compile-verified
//
#include <hip/hip_runtime.h>
#include <hip/hip_bf16.h>

typedef __attribute__((ext_vector_type(2))) float v2f;
typedef __attribute__((ext_vector_type(8))) float v8f;

#define FLTMAX 3.402823466e+38f

// Problem constants (from reference): b=8, c=3, h=w=64 -> q=4096, p=2048
#define NB 8
#define NP 2048
#define NQ 4096

// ---------------------------------------------------------------------------
// Prep over (b,q): ypad planes (y0,y1,y2,0), wy2 = sum_c w2*y^2, colmin=+inf
// ---------------------------------------------------------------------------
__global__ void chamfer_prep_y(const float* __restrict__ y,
                               const float* __restrict__ w,
                               float* __restrict__ ypad,
                               float* __restrict__ wy2,
                               float* __restrict__ colmin) {
    int i = blockIdx.x * blockDim.x + threadIdx.x;   // 0 .. NB*NQ-1
    if (i >= NB * NQ) return;
    int b = i >> 12;
    int q = i & (NQ - 1);
    float s0 = w[0] * w[0], s1 = w[1] * w[1], s2 = w[2] * w[2];
    float y0 = y[(b * 3 + 0) * NQ + q];
    float y1 = y[(b * 3 + 1) * NQ + q];
    float y2 = y[(b * 3 + 2) * NQ + q];
    ypad[(b * 4 + 0) * NQ + q] = y0;
    ypad[(b * 4 + 1) * NQ + q] = y1;
    ypad[(b * 4 + 2) * NQ + q] = y2;
    ypad[(b * 4 + 3) * NQ + q] = 0.0f;
    wy2[i] = s0 * y0 * y0 + s1 * y1 * y1 + s2 * y2 * y2;
    colmin[i] = FLTMAX;
}

// ---------------------------------------------------------------------------
// Prep over (b,p): tw4 = (t*w2, pad 0), wt2 = sum_c w2*t^2, rowmin=+inf
// ---------------------------------------------------------------------------
__global__ void chamfer_prep_t(const float* __restrict__ t,
                               const float* __restrict__ w,
                               float* __restrict__ tw4,
                               float* __restrict__ wt2,
                               float* __restrict__ rowmin) {
    int i = blockIdx.x * blockDim.x + threadIdx.x;   // 0 .. NB*NP-1
    if (i >= NB * NP) return;
    float s0 = w[0] * w[0], s1 = w[1] * w[1], s2 = w[2] * w[2];
    float t0 = t[i * 3 + 0];
    float t1 = t[i * 3 + 1];
    float t2 = t[i * 3 + 2];
    tw4[i * 4 + 0] = t0 * s0;
    tw4[i * 4 + 1] = t1 * s1;
    tw4[i * 4 + 2] = t2 * s2;
    tw4[i * 4 + 3] = 0.0f;
    wt2[i] = s0 * t0 * t0 + s1 * t1 * t1 + s2 * t2 * t2;
    rowmin[i] = FLTMAX;
}

// ---------------------------------------------------------------------------
// Main: WMMA f32 16x16x4 tiles of cross = (t*w2) x y, fused min reductions.
// Grid: NB * 8(p-blocks) * 2(q-halves) = 128 blocks, 512 threads (16 waves).
// Each wave: 16 p-rows x 2048 q-range = 128 tiles, 1 WMMA per tile.
// Column mins: both lane-halves issue ds_min_num_f32 to the same 16 LDS
// addresses (DS unit merges same-address atomics) -- no shuffle, no branch.
// ---------------------------------------------------------------------------
__global__ __launch_bounds__(512) void chamfer_main(
    const float* __restrict__ ypad, const float* __restrict__ wy2,
    const float* __restrict__ tw4,  const float* __restrict__ wt2,
    float* __restrict__ colmin,     float* __restrict__ rowmin) {
    __shared__ float cmin_s[2048];

    const int bidx = blockIdx.x;
    const int b    = bidx >> 4;
    const int pblk = (bidx >> 1) & 7;
    const int qblk = bidx & 1;
    const int tid  = threadIdx.x;

    for (int i = tid; i < 2048; i += 512) cmin_s[i] = FLTMAX;
    __syncthreads();

    const int wave = tid >> 5;
    const int lane = tid & 31;
    const int half = lane >> 4;        // 0: lanes 0-15, 1: lanes 16-31
    const int l16  = lane & 15;
    const int p0   = pblk * 256 + wave * 16;
    const int qbase = qblk * 2048;

    // A matrix (16x4 f32): VGPR0 = {K0 | K2}, VGPR1 = {K1 | K3} per half.
    // tw4 row layout (tw0,tw1,tw2,0) -> half 0 reads (tw0,tw1), half 1 (tw2,0).
    const float* arow = tw4 + ((size_t)(b * NP + p0 + l16)) * 4 + half * 2;
    v2f a;
    a.x = arow[0];
    a.y = arow[1];

    // wt2 values for the 16 rows: D VGPR r holds row p0 + r + 8*half.
    float wt2v[8];
#pragma unroll
    for (int r = 0; r < 8; ++r) wt2v[r] = wt2[b * NP + p0 + half * 8 + r];

    // B matrix planes: half 0 -> (y0, y1), half 1 -> (y2, zero-pad plane).
    const float* yb0  = ypad + ((size_t)(b * 4 + half * 2)) * NQ;
    const float* yb1  = yb0 + NQ;
    const float* wy2b = wy2 + b * NQ;

    v8f rmin;
#pragma unroll
    for (int r = 0; r < 8; ++r) rmin[r] = FLTMAX;

    for (int qt = 0; qt < 2048; qt += 16) {
        const int q = qbase + qt + l16;
        v2f bm;
        bm.x = yb0[q];
        bm.y = yb1[q];
        float wy2l = wy2b[q];

        v8f c = {};
        c = __builtin_amdgcn_wmma_f32_16x16x4_f32(
            /*neg_a=*/false, a, /*neg_b=*/false, bm,
            /*c_mod=*/(short)0, c, /*reuse_a=*/false, /*reuse_b=*/false);

        float cm = FLTMAX;
#pragma unroll
        for (int r = 0; r < 8; ++r) {
            float m = fmaf(-2.0f, c[r], wt2v[r] + wy2l);
            rmin[r] = fminf(rmin[r], m);
            cm      = fminf(cm, m);
        }
        // cm covers this half's 8 rows; the other half's ds_min to the same
        // address completes the 16-row column min.
        atomicMin(&cmin_s[qt + l16], cm);
    }

    // Row min: reduce across the 16 lanes of each half (q direction).
#pragma unroll
    for (int r = 0; r < 8; ++r) {
        float v = rmin[r];
        v = fminf(v, __shfl_xor(v, 8));
        v = fminf(v, __shfl_xor(v, 4));
        v = fminf(v, __shfl_xor(v, 2));
        v = fminf(v, __shfl_xor(v, 1));
        if (l16 == 0) atomicMin(&rowmin[b * NP + p0 + half * 8 + r], v);
    }

    __syncthreads();
    for (int i = tid; i < 2048; i += 512)
        atomicMin(&colmin[b * NQ + qbase + i], cmin_s[i]);
}

// ---------------------------------------------------------------------------
// Final: out = mean(rowmin) + mean(colmin), single block.
// ---------------------------------------------------------------------------
__global__ __launch_bounds__(1024) void chamfer_reduce(
    const float* __restrict__ rowmin, const float* __restrict__ colmin,
    float* __restrict__ out) {
    __shared__ float sdata[1024];
    const int tid = threadIdx.x;
    float s = 0.0f;
    for (int i = tid; i < NB * NP; i += 1024)
        s += rowmin[i] * (1.0f / (float)(NB * NP));
    for (int i = tid; i < NB * NQ; i += 1024)
        s += colmin[i] * (1.0f / (float)(NB * NQ));
    sdata[tid] = s;
    __syncthreads();
    for (int off = 512; off > 0; off >>= 1) {
        if (tid < off) sdata[tid] += sdata[tid + off];
        __syncthreads();
    }
    if (tid == 0) out[0] = sdata[0];
}

// ---------------------------------------------------------------------------
extern "C" void kernel_launch(void* const* d_in, const int* in_sizes, int n_in,
                              void* d_out, int out_size, void* d_ws, size_t ws_size,
                              hipStream_t stream) {
    const float* y = (const float*)d_in[0];   // [8,3,64,64]
    const float* t = (const float*)d_in[1];   // [8,2048,3]
    const float* w = (const float*)d_in[2];   // [3]
    float* out = (float*)d_out;

    // Workspace layout (floats)
    float* ws     = (float*)d_ws;
    float* ypad   = ws;                         // NB*4*NQ   = 131072
    float* wy2    = ypad + NB * 4 * NQ;         // NB*NQ     =  32768
    float* tw4    = wy2 + NB * NQ;              // NB*NP*4   =  65536
    float* wt2    = tw4 + NB * NP * 4;          // NB*NP     =  16384
    float* colmin = wt2 + NB * NP;              // NB*NQ     =  32768
    float* rowmin = colmin + NB * NQ;           // NB*NP     =  16384

    chamfer_prep_y<<<(NB * NQ + 255) / 256, 256, 0, stream>>>(y, w, ypad, wy2, colmin);
    chamfer_prep_t<<<(NB * NP + 255) / 256, 256, 0, stream>>>(t, w, tw4, wt2, rowmin);
    chamfer_main<<<NB * 8 * 2, 512, 0, stream>>>(ypad, wy2, tw4, wt2, colmin, rowmin);
    chamfer_reduce<<<1, 1024, 0, stream>>>(rowmin, colmin, out);
}